// contracting_REN_317827580427
// MI455X (gfx1250) — compile-verified
//
#include <hip/hip_runtime.h>
#include <hip/hip_bf16.h>

#define S_DIM   8
#define N_DIM   64
#define SN_DIM  80           // 2*S + N
#define EPSV    1.0e-4f
#define WS_WV   0            // [64][16]  : row j = [C1[j,:]/L, calD12[j,:]/L, bw/L, 0]
#define WS_DTT  1024         // [64][64]  : DtT[i][j] = Dtilde[j][i] (j>i else 0)
#define WS_WO   (1024+4096)  // [13][80]  : row o = [Wx(8), Wu(6), bias, 0, Wn(64)] (o<8 pre-multiplied by Einv)

typedef float v2f __attribute__((ext_vector_type(2)));
typedef float v8f __attribute__((ext_vector_type(8)));

__device__ __forceinline__ float ftanh(float x) {
#if __has_builtin(__builtin_amdgcn_tanhf)
    return __builtin_amdgcn_tanhf(x);
#else
    float r;
    // gfx1250 VOP1 transcendental; v_nop covers the TRANS-op RAW hazard
    asm volatile("v_tanh_f32 %0, %1\n\tv_nop" : "=v"(r) : "v"(x));
    return r;
#endif
}

// ---------------------------------------------------------------------------
// Kernel 1: H = X^T X + eps*I via V_WMMA_F32_16X16X4_F32 (5x5 tiles, 25 waves),
// then fold all small matrices into the 3 weight tables used by the batch pass.
// ---------------------------------------------------------------------------
__global__ void __launch_bounds__(1024)
prep_kernel(const float* __restrict__ X,      const float* __restrict__ calB2,
            const float* __restrict__ C2,     const float* __restrict__ calD12,
            const float* __restrict__ D21,    const float* __restrict__ D22,
            const float* __restrict__ Y1,     const float* __restrict__ biasvec,
            float* __restrict__ ws)
{
    __shared__ float Hs[SN_DIM * SN_DIM];
    __shared__ float sE[64];
    __shared__ float sI[64];
    __shared__ float sLam[64];

    const int tid  = threadIdx.x;
    const int wv   = tid >> 5;
    const int lane = tid & 31;

    if (wv < 25) {
        const int tm   = wv / 5;
        const int tn   = wv % 5;
        const int half = lane >> 4;       // K sub-pair select
        const int l    = lane & 15;       // M (for A) / N (for B)
        const int rowA = tm * 16 + l;
        const int rowB = tn * 16 + l;

        v8f c = {};
        for (int k0 = 0; k0 < SN_DIM; k0 += 4) {
            const int ka = k0 + 2 * half;
            v2f a, b;
            // A[m][k] = X[k][m] ; 16x4 f32 A layout: VGPR0:K={0|2}, VGPR1:K={1|3}
            a.x = X[(ka    ) * SN_DIM + rowA];
            a.y = X[(ka + 1) * SN_DIM + rowA];
            // B[k][n] = X[k][n] ; assume mirrored K striping for the 4x16 B tile
            b.x = X[(ka    ) * SN_DIM + rowB];
            b.y = X[(ka + 1) * SN_DIM + rowB];
            c = __builtin_amdgcn_wmma_f32_16x16x4_f32(
                    false, a, false, b, (short)0, c, false, false);
        }
        // C/D layout: VGPR r -> M = r + 8*half, lane l -> N
        #pragma unroll
        for (int r = 0; r < 8; ++r) {
            const int gr = tm * 16 + r + 8 * half;
            const int gc = tn * 16 + l;
            float hv = c[r];
            if (gr == gc) hv += EPSV;
            Hs[gr * SN_DIM + gc] = hv;
        }
    }
    __syncthreads();

    // E = 0.5*(H11 + calP + Y1);  Lambda = 0.5*diag(H22)
    if (tid < 64) {
        const int i = tid >> 3, j = tid & 7;
        sE[tid]   = 0.5f * (Hs[i * SN_DIM + j] + Hs[(72 + i) * SN_DIM + 72 + j] + Y1[i * 8 + j]);
        sI[tid]   = (i == j) ? 1.0f : 0.0f;
        sLam[tid] = 0.5f * Hs[(8 + tid) * SN_DIM + 8 + tid];
    }
    __syncthreads();

    // 8x8 Gauss-Jordan inverse of E -> sI (single thread; trivial cost)
    if (tid == 0) {
        for (int col = 0; col < 8; ++col) {
            int p = col; float best = fabsf(sE[col * 8 + col]);
            for (int r = col + 1; r < 8; ++r) {
                float av = fabsf(sE[r * 8 + col]);
                if (av > best) { best = av; p = r; }
            }
            if (p != col) {
                for (int k = 0; k < 8; ++k) {
                    float t = sE[col * 8 + k]; sE[col * 8 + k] = sE[p * 8 + k]; sE[p * 8 + k] = t;
                    t = sI[col * 8 + k]; sI[col * 8 + k] = sI[p * 8 + k]; sI[p * 8 + k] = t;
                }
            }
            const float inv = 1.0f / sE[col * 8 + col];
            for (int k = 0; k < 8; ++k) { sE[col * 8 + k] *= inv; sI[col * 8 + k] *= inv; }
            for (int r = 0; r < 8; ++r) if (r != col) {
                const float f = sE[r * 8 + col];
                for (int k = 0; k < 8; ++k) {
                    sE[r * 8 + k] -= f * sE[col * 8 + k];
                    sI[r * 8 + k] -= f * sI[col * 8 + k];
                }
            }
        }
    }
    __syncthreads();

    float* __restrict__ Wv  = ws + WS_WV;
    float* __restrict__ DtT = ws + WS_DTT;
    float* __restrict__ Wo  = ws + WS_WO;

    // Wv: row j = [ C1[j,:]/L_j (8) | calD12[j,:]/L_j (6) | bw_j/L_j | 0 ]
    if (tid < 64) {
        const int j = tid;
        const float inv = 1.0f / sLam[j];
        float* wr = Wv + j * 16;
        for (int k = 0; k < 8; ++k) wr[k]     = -Hs[(8 + j) * SN_DIM + k] * inv;  // C1 = -H[s:s+n,:s]
        for (int k = 0; k < 6; ++k) wr[8 + k] = calD12[j * 6 + k] * inv;
        wr[14] = biasvec[8 + j] * inv;
        wr[15] = 0.0f;
    }

    // DtT[i][j] = Dtilde[j][i] = D11[j][i]/L_j = (j>i) ? -H22[j][i]/L_j : 0
    for (int t = tid; t < 64 * 64; t += blockDim.x) {
        const int i = t >> 6, j = t & 63;
        DtT[t] = (j > i) ? (-Hs[(8 + j) * SN_DIM + 8 + i] / sLam[j]) : 0.0f;
    }

    // Wo: rows 0..7 pre-multiplied by Einv (folds hidden = Ehidden @ Einv.T)
    for (int t = tid; t < 13 * 80; t += blockDim.x) {
        const int o = t / 80, cidx = t % 80;
        float val = 0.0f;
        if (o < 8) {
            if (cidx < 8) {        // Einv @ F_
                for (int k = 0; k < 8; ++k) val += sI[o * 8 + k] * Hs[(72 + k) * SN_DIM + cidx];
            } else if (cidx < 14) { // Einv @ calB_2
                for (int k = 0; k < 8; ++k) val += sI[o * 8 + k] * calB2[k * 6 + (cidx - 8)];
            } else if (cidx == 14) { // Einv @ bx
                for (int k = 0; k < 8; ++k) val += sI[o * 8 + k] * biasvec[k];
            } else if (cidx == 15) {
                val = 0.0f;
            } else {               // Einv @ B1
                for (int k = 0; k < 8; ++k) val += sI[o * 8 + k] * Hs[(72 + k) * SN_DIM + 8 + (cidx - 16)];
            }
        } else {
            const int m = o - 8;
            if (cidx < 8)        val = C2[m * 8 + cidx];
            else if (cidx < 14)  val = D22[m * 6 + (cidx - 8)];
            else if (cidx == 14) val = biasvec[72 + m];
            else if (cidx == 15) val = 0.0f;
            else                 val = D21[m * 64 + (cidx - 16)];
        }
        Wo[t] = val;
    }
}

// ---------------------------------------------------------------------------
// Kernel 2: one lane per batch row (wave32 -> 32 rows/wave), v[64] in VGPRs,
// fully unrolled triangular scan with uniform (scalar-load) weights.
// ---------------------------------------------------------------------------
__global__ void __launch_bounds__(256)
ren_batch_kernel(const float* __restrict__ hs, const float* __restrict__ uu,
                 const float* __restrict__ ws, float* __restrict__ out, int B)
{
    const int row = blockIdx.x * blockDim.x + threadIdx.x;
    if (row >= B) return;

    const float* __restrict__ Wv  = ws + WS_WV;
    const float* __restrict__ DtT = ws + WS_DTT;
    const float* __restrict__ Wo  = ws + WS_WO;

    const float4* xp = (const float4*)(hs + (size_t)row * 8);
    const float4 x0 = xp[0], x1 = xp[1];
    float xv[8] = {x0.x, x0.y, x0.z, x0.w, x1.x, x1.y, x1.z, x1.w};

    const float2* up = (const float2*)(uu + (size_t)row * 6);
    const float2 u0 = up[0], u1 = up[1], u2 = up[2];
    float uv[6] = {u0.x, u0.y, u1.x, u1.y, u2.x, u2.y};

    // v0 = (x @ C1.T + u @ calD12.T + bw) / Lambda   (all folded into Wv)
    float v[64];
    #pragma unroll
    for (int j = 0; j < 64; ++j) {
        const float* wr = Wv + j * 16;
        float a = wr[14];
        #pragma unroll
        for (int k = 0; k < 8; ++k) a = fmaf(xv[k], wr[k], a);
        #pragma unroll
        for (int k = 0; k < 6; ++k) a = fmaf(uv[k], wr[8 + k], a);
        v[j] = a;
    }

    // Sequential triangular scan; v[i] is final at step i, so w[i] = tanh(v[i])
    #pragma unroll
    for (int i = 0; i < 64; ++i) {
        const float t = ftanh(v[i]);
        const float* dr = DtT + i * 64;
        #pragma unroll
        for (int j = i + 1; j < 64; ++j) v[j] = fmaf(t, dr[j], v[j]);
        v[i] = t;               // v[] now holds w[] progressively
    }

    // outputs = [x, u, 1, w] @ Wo.T   (rows 0..7 = hidden, 8..12 = y)
    float res[13];
    #pragma unroll
    for (int o = 0; o < 13; ++o) {
        const float* wr = Wo + o * 80;
        float a = wr[14];
        #pragma unroll
        for (int k = 0; k < 8; ++k) a = fmaf(xv[k], wr[k], a);
        #pragma unroll
        for (int k = 0; k < 6; ++k) a = fmaf(uv[k], wr[8 + k], a);
        #pragma unroll
        for (int j = 0; j < 64; ++j) a = fmaf(v[j], wr[16 + j], a);
        res[o] = a;
    }

    float4* ho = (float4*)(out + (size_t)row * 8);
    ho[0] = make_float4(res[0], res[1], res[2], res[3]);
    ho[1] = make_float4(res[4], res[5], res[6], res[7]);
    float* yo = out + (size_t)B * 8 + (size_t)row * 5;
    yo[0] = res[8]; yo[1] = res[9]; yo[2] = res[10]; yo[3] = res[11]; yo[4] = res[12];
}

extern "C" void kernel_launch(void* const* d_in, const int* in_sizes, int n_in,
                              void* d_out, int out_size, void* d_ws, size_t ws_size,
                              hipStream_t stream) {
    const float* hs      = (const float*)d_in[0];
    const float* uu      = (const float*)d_in[1];
    const float* X       = (const float*)d_in[2];
    const float* calB2   = (const float*)d_in[3];
    const float* C2      = (const float*)d_in[4];
    const float* calD12  = (const float*)d_in[5];
    const float* D21     = (const float*)d_in[6];
    const float* D22     = (const float*)d_in[7];
    const float* Y1      = (const float*)d_in[8];
    const float* biasvec = (const float*)d_in[9];
    float*       out     = (float*)d_out;
    float*       ws      = (float*)d_ws;

    const int B = in_sizes[0] / 8;

    prep_kernel<<<1, 1024, 0, stream>>>(X, calB2, C2, calD12, D21, D22, Y1, biasvec, ws);

    const int threads = 256;
    const int blocks  = (B + threads - 1) / threads;
    ren_batch_kernel<<<blocks, threads, 0, stream>>>(hs, uu, ws, out, B);
}